// LengthRegulator_5153960755461
// MI455X (gfx1250) — compile-verified
//
#include <hip/hip_runtime.h>

// Length regulator: out[b, j, :] = enc[b, searchsorted(cumsum(max(dur,1)), j, 'right'), :]
// zero-padded past total length. Shapes fixed by the reference:
//   enc: [32, 512, 512] f32, dur: [32, 512] i32, t_out = 4096.
// Pure bandwidth problem: 256 MB NT stores + 32 MB L2-resident loads -> ~13 us floor @ 23.3 TB/s.

typedef __attribute__((ext_vector_type(4))) float v4f;

namespace {
constexpr int kB    = 32;
constexpr int kT    = 512;
constexpr int kD    = 512;
constexpr int kTout = 4096;
constexpr int kFramesPerBlock = 8;                 // one wave32 per output frame
constexpr int kThreads        = kFramesPerBlock * 32;
constexpr int kVecPerLane     = kD / (4 * 32);     // 512 floats / (float4 * 32 lanes) = 4
}

// One wave32 per batch row: inclusive scan of max(dur,1) via lane shuffles.
__global__ __launch_bounds__(32) void lr_cumsum_kernel(const int* __restrict__ dur,
                                                       int* __restrict__ ends) {
    const int b    = blockIdx.x;
    const int lane = threadIdx.x;                  // blockDim.x == 32 (single wave)
    int carry = 0;
    for (int base = 0; base < kT; base += 32) {
        int v = dur[b * kT + base + lane];
        v = (v > 1) ? v : 1;
        #pragma unroll
        for (int off = 1; off < 32; off <<= 1) {   // wave32 inclusive scan
            int n = __shfl_up(v, off, 32);
            if (lane >= off) v += n;
        }
        v += carry;
        ends[b * kT + base + lane] = v;
        carry = __shfl(v, 31, 32);                 // broadcast chunk total
    }
}

// 8 wave32s per block; each wave binary-searches the LDS-resident ends[] table
// for its frame, then streams one 2 KB encoder row with coalesced b128 ops.
__global__ __launch_bounds__(kThreads) void lr_gather_kernel(const float* __restrict__ enc,
                                                             const int* __restrict__ ends,
                                                             float* __restrict__ out) {
    __shared__ int s_ends[kT];

    constexpr int blocksPerBatch = kTout / kFramesPerBlock;   // 512
    const int b   = blockIdx.x / blocksPerBatch;
    const int j0  = (blockIdx.x % blocksPerBatch) * kFramesPerBlock;
    const int tid = threadIdx.x;

    // Stage the 2 KB cumulative-ends table for this batch into LDS.
    for (int i = tid; i < kT; i += kThreads) s_ends[i] = ends[b * kT + i];
    __syncthreads();

    const int total = s_ends[kT - 1];
    const int wave  = tid >> 5;
    const int lane  = tid & 31;
    const int j     = j0 + wave;                   // output frame owned by this wave

    // searchsorted(ends, j, side='right'): first t with ends[t] > j.
    int lo = 0, hi = kT;
    while (lo < hi) {
        const int mid = (lo + hi) >> 1;
        if (s_ends[mid] <= j) lo = mid + 1; else hi = mid;
    }
    const int idx = (lo < kT - 1) ? lo : (kT - 1);

    v4f* dst = (v4f*)(out + ((size_t)b * kTout + j) * kD);
    if (j < total) {
        const v4f* src = (const v4f*)(enc + ((size_t)b * kT + idx) * kD);
        #pragma unroll
        for (int k = 0; k < kVecPerLane; ++k) {
            // Load stays regular-temporal (encoder reused by neighboring frames, L2-resident);
            // store is non-temporal (write-once 256 MB stream, bigger than L2).
            v4f v = src[lane + 32 * k];
            __builtin_nontemporal_store(v, dst + lane + 32 * k);
        }
    } else {
        const v4f z = (v4f)0.0f;                   // zero-pad tail frames
        #pragma unroll
        for (int k = 0; k < kVecPerLane; ++k)
            __builtin_nontemporal_store(z, dst + lane + 32 * k);
    }
}

extern "C" void kernel_launch(void* const* d_in, const int* in_sizes, int n_in,
                              void* d_out, int out_size, void* d_ws, size_t ws_size,
                              hipStream_t stream) {
    (void)in_sizes; (void)n_in; (void)out_size; (void)ws_size;
    const float* enc = (const float*)d_in[0];      // [32,512,512] f32
    const int*   dur = (const int*)d_in[1];        // [32,512] i32
    float*       out = (float*)d_out;              // [32,4096,512] f32
    int*         ends = (int*)d_ws;                // scratch: 32*512 i32 = 64 KB

    lr_cumsum_kernel<<<kB, 32, 0, stream>>>(dur, ends);
    lr_gather_kernel<<<kB * (kTout / kFramesPerBlock), kThreads, 0, stream>>>(enc, ends, out);
}